// MultiLIF_17059610100026
// MI455X (gfx1250) — compile-verified
//
#include <hip/hip_runtime.h>

// Adaptive-LIF forward scan, MI455X (gfx1250, wave32).
// Memory-bound streaming kernel: 64 MiB in, 192 MiB out, ~11.5 us roofline
// at 23.3 TB/s. No matrix structure -> no WMMA; the CDNA5-specific paths used
// are global_prefetch_b8 and non-temporal (TH=NT) wide stores.

typedef __attribute__((ext_vector_type(2))) float f32x2;

namespace {
constexpr int kB = 32;
constexpr int kL = 1024;
constexpr int kK = 512;
constexpr int kVec = 2;                              // f32x2 per thread
constexpr int kThreadsPerB   = kK / kVec;            // 256
constexpr int kTotalThreads  = kB * kThreadsPerB;    // 8192 lanes = 256 waves
constexpr int kBlock = 32;                           // one wave32 per workgroup
constexpr int kGrid  = kTotalThreads / kBlock;       // 256 workgroups -> max WGP spread
constexpr int kPrefetchT = 16;                       // prefetch 16 timesteps (32 KB/row) ahead

constexpr float kInvTau      = 1.0f / 20.0f;    // 1/TAU
constexpr float kTh          = 1.5f;            // TH
constexpr float kInvTauAdapt = 1.0f / 100.0f;   // 1/TAU_ADAPT
constexpr float kBeta        = 1.5f;            // BETA_ADAPT
constexpr float kVReset      = -0.5f;           // V_RESET
} // namespace

__global__ __launch_bounds__(kBlock) void multi_lif_scan(
    const float* __restrict__ I,
    float* __restrict__ spikes,
    float* __restrict__ series,
    float* __restrict__ vseq)
{
    const int gid = blockIdx.x * kBlock + threadIdx.x;   // 0..8191
    const int b   = gid / kThreadsPerB;
    const int kv  = gid - b * kThreadsPerB;
    // element offset of (b, t=0, k = kv*kVec); t-stride is kK elements (2 KB)
    const size_t base = (size_t)b * kL * kK + (size_t)kv * kVec;
    const float* Irow = I + base;

    f32x2 v    = {0.0f, 0.0f};
    f32x2 a    = {0.0f, 0.0f};
    f32x2 snum = {0.0f, 0.0f};

    // Register double-buffer: I_t already in flight when step t executes.
    f32x2 icur = *(const f32x2*)(Irow);

#pragma unroll 4
    for (int t = 0; t < kL; ++t) {
        // Issue next-step load before this step's stores (restrict => movable).
        const int tn = (t + 1 < kL) ? (t + 1) : (kL - 1);
        f32x2 inext = *(const f32x2*)(Irow + (size_t)tn * kK);

        // gfx1250 global_prefetch_b8: pull the line for t+16 toward L2/WGP$.
        const int tp = (t + kPrefetchT < kL) ? (t + kPrefetchT) : (kL - 1);
        __builtin_prefetch(Irow + (size_t)tp * kK, 0, 1);

        // th_eff = TH + BETA * a
        f32x2 th;
        th.x = fmaf(kBeta, a.x, kTh);
        th.y = fmaf(kBeta, a.y, kTh);

        // v = v + (-v/TAU + I_t)
        v.x = v.x + fmaf(-kInvTau, v.x, icur.x);
        v.y = v.y + fmaf(-kInvTau, v.y, icur.y);

        // Forward value of the STE spike is exactly s_hard (see analysis):
        // s_hard==1 implies s_soft>=0.5 so (1-s_soft)+s_soft == 1.0 exactly;
        // s_hard==0 gives (0-s)+s == 0.0 exactly. No sigmoid needed.
        const bool fx = (v.x >= th.x);
        const bool fy = (v.y >= th.y);
        f32x2 s = { fx ? 1.0f : 0.0f, fy ? 1.0f : 0.0f };
        snum.x += s.x;
        snum.y += s.y;

        // Streaming outputs: written once, never re-read -> non-temporal.
        const size_t o = base + (size_t)t * kK;
        __builtin_nontemporal_store(s,    (f32x2*)(spikes + o));   // s at step t
        __builtin_nontemporal_store(snum, (f32x2*)(series + o));   // s_num AFTER add
        __builtin_nontemporal_store(v,    (f32x2*)(vseq   + o));   // v BEFORE reset

        // v_next = v*(1-s) + V_RESET*s  with s in {0,1}  ==  select
        v.x = fx ? kVReset : v.x;
        v.y = fy ? kVReset : v.y;

        // a_next = a + (-a/TAU_ADAPT + s)
        a.x = a.x + fmaf(-kInvTauAdapt, a.x, s.x);
        a.y = a.y + fmaf(-kInvTauAdapt, a.y, s.y);

        icur = inext;
    }
}

extern "C" void kernel_launch(void* const* d_in, const int* in_sizes, int n_in,
                              void* d_out, int out_size, void* d_ws, size_t ws_size,
                              hipStream_t stream) {
    (void)in_sizes; (void)n_in; (void)out_size; (void)d_ws; (void)ws_size;
    const float* I = (const float*)d_in[0];
    float* out = (float*)d_out;
    const size_t N = (size_t)kB * kL * kK;  // 16,777,216 elements per output
    // d_out layout: [spikes | series | v_seq], each N floats, in return order.
    multi_lif_scan<<<kGrid, kBlock, 0, stream>>>(I, out, out + N, out + 2 * N);
}